// MoELayer_44702019617359
// MI455X (gfx1250) — compile-verified
//
#include <hip/hip_runtime.h>
#include <hip/hip_bf16.h>
#include <math.h>

// ---------------- problem constants ----------------
#define T_TOK 4096
#define DIM   1024
#define FFN   4096
#define NEXP  8
#define TOPK  2
#define LBW   0.01f

// ---------------- ffn tiling ----------------
#define TILE_M   16
#define FB       512                  // F-slice per block (grid.y = FFN/FB = 8)
#define KC       64                   // K staging chunk for phase 1
#define MAX_ROWS (T_TOK*TOPK + NEXP*TILE_M)   // 8320 padded routed rows
#define MAX_TILES (MAX_ROWS / TILE_M)         // 520
#define XPITCH   (KC + 4)             // 68: bank-conflict-free LDS pitch
#define HPITCH   (FB + 4)             // 516

// gfx1250 async global->LDS path, guarded so absence degrades gracefully
#if defined(__has_builtin)
#if __has_builtin(__builtin_amdgcn_global_load_async_to_lds_b128) && \
    __has_builtin(__builtin_amdgcn_s_wait_asynccnt)
#define USE_ASYNC_LDS 1
#endif
#endif

typedef float v2f __attribute__((ext_vector_type(2)));
typedef float v8f __attribute__((ext_vector_type(8)));
typedef int   v4i __attribute__((ext_vector_type(4)));
typedef __attribute__((address_space(1))) v4i glb_v4i;   // global (AS1)
typedef __attribute__((address_space(3))) v4i lds_v4i;   // LDS (AS3)

// =====================================================================
// 1) Router: logits -> softmax(8) -> top-2 -> normalized gates
// =====================================================================
__global__ void moe_router_kernel(const float* __restrict__ x,
                                  const float* __restrict__ Wr,
                                  int* __restrict__ topi, float* __restrict__ topw,
                                  float* __restrict__ probs_sum,
                                  int* __restrict__ expert_count) {
    int wave = threadIdx.x >> 5;
    int lane = threadIdx.x & 31;
    int t = blockIdx.x * 8 + wave;          // 8 waves/block, 512 blocks
    if (t >= T_TOK) return;
    const float* xr = x + (size_t)t * DIM;
    float acc[NEXP];
#pragma unroll
    for (int e = 0; e < NEXP; ++e) acc[e] = 0.f;
    for (int d = lane; d < DIM; d += 32) {
        float xv = xr[d];
        const float* wr = Wr + (size_t)d * NEXP;
#pragma unroll
        for (int e = 0; e < NEXP; ++e) acc[e] += xv * wr[e];
    }
#pragma unroll
    for (int e = 0; e < NEXP; ++e)
        for (int off = 16; off > 0; off >>= 1)
            acc[e] += __shfl_down(acc[e], off, 32);
    if (lane == 0) {
        float m = acc[0];
#pragma unroll
        for (int e = 1; e < NEXP; ++e) m = fmaxf(m, acc[e]);
        float p[NEXP], s = 0.f;
#pragma unroll
        for (int e = 0; e < NEXP; ++e) { p[e] = __expf(acc[e] - m); s += p[e]; }
        float inv = 1.f / s;
#pragma unroll
        for (int e = 0; e < NEXP; ++e) { p[e] *= inv; atomicAdd(&probs_sum[e], p[e]); }
        int i0 = 0;
#pragma unroll
        for (int e = 1; e < NEXP; ++e) if (p[e] > p[i0]) i0 = e;
        int i1 = (i0 == 0) ? 1 : 0;
#pragma unroll
        for (int e = 0; e < NEXP; ++e) if (e != i0 && p[e] > p[i1]) i1 = e;
        float w0 = p[i0], w1 = p[i1], sw = 1.f / (w0 + w1);
        topi[t * 2 + 0] = i0;  topi[t * 2 + 1] = i1;
        topw[t * 2 + 0] = w0 * sw;  topw[t * 2 + 1] = w1 * sw;
        atomicAdd(&expert_count[i0], 1);
        atomicAdd(&expert_count[i1], 1);
    }
}

// =====================================================================
// 2) tiny helpers: zero / scan+aux / init lists / scatter / out init
// =====================================================================
__global__ void moe_zero_small(int* counts, int* fill, float* psum) {
    int i = threadIdx.x;
    if (i < NEXP) { counts[i] = 0; fill[i] = 0; psum[i] = 0.f; }
}

__global__ void moe_scan_aux(const int* __restrict__ counts,
                             int* __restrict__ offsets,
                             const float* __restrict__ psum,
                             float* __restrict__ aux_out) {
    if (threadIdx.x == 0 && blockIdx.x == 0) {
        int off = 0;
        float aux = 0.f;
        for (int e = 0; e < NEXP; ++e) {
            offsets[e] = off;
            int c = counts[e];
            off += (c + TILE_M - 1) & ~(TILE_M - 1);          // pad to 16
            aux += (psum[e] / (float)T_TOK) * ((float)c / (float)(T_TOK * TOPK));
        }
        offsets[NEXP] = off;
        *aux_out = LBW * (float)NEXP * aux;
    }
}

__global__ void moe_init_lists(int* tok, float* gate) {
    int i = blockIdx.x * blockDim.x + threadIdx.x;
    if (i < MAX_ROWS) { tok[i] = -1; gate[i] = 0.f; }
}

__global__ void moe_scatter(const int* __restrict__ topi, const float* __restrict__ topw,
                            const int* __restrict__ offsets, int* __restrict__ fill,
                            int* __restrict__ tok, float* __restrict__ gate) {
    int i = blockIdx.x * blockDim.x + threadIdx.x;    // over T*K slots
    if (i >= T_TOK * TOPK) return;
    int e = topi[i];
    int pos = atomicAdd(&fill[e], 1);
    int dst = offsets[e] + pos;
    tok[dst]  = i >> 1;
    gate[dst] = topw[i];
}

__global__ void moe_out_init(const int* __restrict__ topi, const float* __restrict__ topw,
                             const float* __restrict__ b2, float* __restrict__ out) {
    int i = blockIdx.x * blockDim.x + threadIdx.x;
    if (i >= T_TOK * DIM) return;
    int t = i / DIM, d = i - t * DIM;
    int e0 = topi[t * 2], e1 = topi[t * 2 + 1];
    out[i] = topw[t * 2] * b2[e0 * DIM + d] + topw[t * 2 + 1] * b2[e1 * DIM + d];
}

// =====================================================================
// 3) Fused expert FFN on fp32 WMMA: gelu(x@W1+b1)@W2, gated scatter-add
//    grid = (MAX_TILES, FFN/FB), block = 256 (8 waves)
// =====================================================================
__global__ __launch_bounds__(256)
void moe_ffn_kernel(const float* __restrict__ x,
                    const float* __restrict__ W1, const float* __restrict__ b1,
                    const float* __restrict__ W2,
                    const int* __restrict__ offsets,
                    const int* __restrict__ tok_list, const float* __restrict__ gate_list,
                    float* __restrict__ out) {
    __shared__ __align__(16) float xs[TILE_M * XPITCH];   // x K-chunk
    __shared__ __align__(16) float hs[TILE_M * HPITCH];   // gelu(xW1+b1) slice
    __shared__ int   po_s[NEXP + 1];
    __shared__ int   toks_s[TILE_M];
    __shared__ float gates_s[TILE_M];

    if (threadIdx.x <= NEXP) po_s[threadIdx.x] = offsets[threadIdx.x];
    __syncthreads();

    const int row0 = blockIdx.x * TILE_M;
    if (row0 >= po_s[NEXP]) return;                       // uniform exit: past routed rows
    int e = 0;
    while (e < NEXP - 1 && row0 >= po_s[e + 1]) ++e;

    if (threadIdx.x < TILE_M) {
        toks_s[threadIdx.x]  = tok_list[row0 + threadIdx.x];
        gates_s[threadIdx.x] = gate_list[row0 + threadIdx.x];
    }
    __syncthreads();

    const int wave = threadIdx.x >> 5;
    const int lane = threadIdx.x & 31;
    const int mrow = lane & 15;            // A/B column-lane role
    const int koff = (lane >> 4) * 2;      // K sub-offset per half-wave
    const int nblk = blockIdx.y * FB;      // F slice base

    const float* W1e = W1 + (size_t)e * DIM * FFN;
    const float* W2e = W2 + (size_t)e * FFN * DIM;

    // ---------------- Phase 1: h = gelu(x @ W1 + b1) for this F slice ----------------
    v8f acc[4];
#pragma unroll
    for (int j = 0; j < 4; ++j) acc[j] = (v8f)0.f;

    for (int k0 = 0; k0 < DIM; k0 += KC) {
        __syncthreads();
        {   // stage x[tile rows][k0 .. k0+63] into LDS (pad rows -> zero)
            int r = threadIdx.x >> 4;
            int c = (threadIdx.x & 15) * 4;
            int t = toks_s[r];
#if defined(USE_ASYNC_LDS)
            if (t >= 0) {
                __builtin_amdgcn_global_load_async_to_lds_b128(
                    (glb_v4i*)(x + (size_t)t * DIM + k0 + c),
                    (lds_v4i*)&xs[r * XPITCH + c],
                    0, 0);
            } else {
                *(float4*)&xs[r * XPITCH + c] = make_float4(0.f, 0.f, 0.f, 0.f);
            }
#else
            float4 v = make_float4(0.f, 0.f, 0.f, 0.f);
            if (t >= 0) v = *(const float4*)(x + (size_t)t * DIM + k0 + c);
            *(float4*)&xs[r * XPITCH + c] = v;
#endif
            // prefetch next K-chunk of the dominant W1 stream toward L2/WGP$
            if (k0 + KC < DIM) {
                const float* pf = W1e + (size_t)(k0 + KC + (threadIdx.x >> 2)) * FFN
                                + nblk + (threadIdx.x & 3) * 128;
                __builtin_prefetch(pf, 0, 1);
            }
        }
#if defined(USE_ASYNC_LDS)
        __builtin_amdgcn_s_wait_asynccnt(0);
#endif
        __syncthreads();
#pragma unroll 4
        for (int kk = 0; kk < KC; kk += 4) {
            v2f a;
            a.x = xs[mrow * XPITCH + kk + koff];
            a.y = xs[mrow * XPITCH + kk + koff + 1];
            const int krow = k0 + kk + koff;
            const float* brow0 = W1e + (size_t)krow * FFN + nblk + mrow;
#pragma unroll
            for (int j = 0; j < 4; ++j) {
                int n = wave * 64 + j * 16;
                v2f b;
                b.x = brow0[n];
                b.y = brow0[FFN + n];
                acc[j] = __builtin_amdgcn_wmma_f32_16x16x4_f32(
                    false, a, false, b, (short)0, acc[j], false, false);
            }
        }
    }

    // bias + exact GELU, park in LDS (C layout: lane&15 = N, M = r + 8*(lane>>4))
#pragma unroll
    for (int j = 0; j < 4; ++j) {
        int nloc = wave * 64 + j * 16;
        float bias = b1[(size_t)e * FFN + nblk + nloc + mrow];
#pragma unroll
        for (int r = 0; r < 8; ++r) {
            float v = acc[j][r] + bias;
            float g = 0.5f * v * (1.0f + erff(v * 0.70710678f));
            int M = r + ((lane >> 4) << 3);
            hs[M * HPITCH + nloc + mrow] = g;
        }
    }
    __syncthreads();

    // ---------------- Phase 2: partial y = h @ W2, gated atomic scatter ----------------
    for (int nt = wave; nt < DIM / 16; nt += 8) {
        const int n = nt * 16;
        v8f c = (v8f)0.f;
#pragma unroll 4
        for (int kk = 0; kk < FB; kk += 4) {
            v2f a;
            a.x = hs[mrow * HPITCH + kk + koff];
            a.y = hs[mrow * HPITCH + kk + koff + 1];
            const int f = nblk + kk + koff;
            const float* brow0 = W2e + (size_t)f * DIM + n + mrow;
            v2f b;
            b.x = brow0[0];
            b.y = brow0[DIM];
            c = __builtin_amdgcn_wmma_f32_16x16x4_f32(
                false, a, false, b, (short)0, c, false, false);
        }
#pragma unroll
        for (int r = 0; r < 8; ++r) {
            int M = r + ((lane >> 4) << 3);
            int t = toks_s[M];
            if (t >= 0)
                atomicAdd(&out[(size_t)t * DIM + n + mrow], gates_s[M] * c[r]);
        }
    }
}

// =====================================================================
// launch
// =====================================================================
extern "C" void kernel_launch(void* const* d_in, const int* in_sizes, int n_in,
                              void* d_out, int out_size, void* d_ws, size_t ws_size,
                              hipStream_t stream) {
    const float* x   = (const float*)d_in[0];
    const float* Wr  = (const float*)d_in[1];
    const float* W1  = (const float*)d_in[2];
    const float* b1  = (const float*)d_in[3];
    const float* W2  = (const float*)d_in[4];
    const float* b2  = (const float*)d_in[5];
    float* out = (float*)d_out;
    float* aux = out + (size_t)T_TOK * DIM;

    char* ws = (char*)d_ws;
    size_t off = 0;
    auto carve = [&](size_t bytes) { size_t o = off; off = (off + bytes + 255) & ~(size_t)255; return o; };
    int*   topi     = (int*)  (ws + carve((size_t)T_TOK * 2 * sizeof(int)));
    float* topw     = (float*)(ws + carve((size_t)T_TOK * 2 * sizeof(float)));
    int*   counts   = (int*)  (ws + carve(NEXP * sizeof(int)));
    int*   offsets  = (int*)  (ws + carve((NEXP + 1) * sizeof(int)));
    int*   fill     = (int*)  (ws + carve(NEXP * sizeof(int)));
    float* psum     = (float*)(ws + carve(NEXP * sizeof(float)));
    int*   tok_list = (int*)  (ws + carve((size_t)MAX_ROWS * sizeof(int)));
    float* gate_lst = (float*)(ws + carve((size_t)MAX_ROWS * sizeof(float)));
    (void)ws_size; (void)in_sizes; (void)n_in; (void)out_size;

    moe_zero_small<<<1, 32, 0, stream>>>(counts, fill, psum);
    moe_router_kernel<<<T_TOK / 8, 256, 0, stream>>>(x, Wr, topi, topw, psum, counts);
    moe_scan_aux<<<1, 32, 0, stream>>>(counts, offsets, psum, aux);
    moe_init_lists<<<(MAX_ROWS + 255) / 256, 256, 0, stream>>>(tok_list, gate_lst);
    moe_scatter<<<(T_TOK * TOPK + 255) / 256, 256, 0, stream>>>(topi, topw, offsets, fill,
                                                                tok_list, gate_lst);
    moe_out_init<<<(T_TOK * DIM + 255) / 256, 256, 0, stream>>>(topi, topw, b2, out);
    dim3 grid(MAX_TILES, FFN / FB);
    moe_ffn_kernel<<<grid, 256, 0, stream>>>(x, W1, b1, W2, offsets, tok_list, gate_lst, out);
}